// MeanAggregator_27324581937606
// MI455X (gfx1250) — compile-verified
//
#include <hip/hip_runtime.h>

// Mean aggregation over sampled neighbor embeddings (GraphSAGE-style):
//   out[n, :] = (1/S) * sum_s embedding[neigh_idx[n, s], :]
// N=100000 nodes, S=10 neighbors, D=128 features, fp32, int64 indices.
//
// Memory-bound (0.25 FLOP/byte). Embedding table (102.4 MB) is L2-resident
// on MI455X (192 MB L2); strategy is pure bandwidth: one node per wave32,
// one b128 per lane per 512B row, dual return paths (VMEM->VGPR + gfx1250
// async DMA -> LDS).

#define SNEIGH  10
#define DFEAT   128
#define WPB     8        // waves per block (256 threads)
#define DIRECT  4        // rows returned directly to VGPRs
#define LDSROWS 6        // rows staged via async-to-LDS engine

__global__ __launch_bounds__(256)
void mean_agg_gather_kernel(const long long* __restrict__ neigh_idx,
                            const float*     __restrict__ emb,
                            float*           __restrict__ out,
                            int N)
{
    // 8 waves * 6 rows * 512 B = 24 KB of the 320 KB WGP LDS
    __shared__ float stage[WPB][LDSROWS][DFEAT];

    const int lane = threadIdx.x & 31;
    const int wave = threadIdx.x >> 5;
    const int node = blockIdx.x * WPB + wave;
    if (node >= N) return;            // whole wave exits together

    // Wave-uniform index fetch: force the scalar path (s_load of 10 x i64).
    const int snode = __builtin_amdgcn_readfirstlane(node);
    const long long* ip = neigh_idx + (long long)snode * SNEIGH;
    long long id[SNEIGH];
#pragma unroll
    for (int s = 0; s < SNEIGH; ++s) id[s] = ip[s];

    const unsigned lbyte = (unsigned)(lane << 4);   // 16 B per lane, 512 B per wave

    // ---- Rows DIRECT..9: issue on the gfx1250 async DMA engine (ASYNCcnt).
    // Each lane copies its 16 B slice of the 512 B row straight into LDS,
    // bypassing the VGPR return path entirely.
#pragma unroll
    for (int s = 0; s < LDSROWS; ++s) {
        unsigned long long ga =
            (unsigned long long)(const void*)(emb + (size_t)id[DIRECT + s] * DFEAT) + lbyte;
        unsigned la = (unsigned)(uintptr_t)(&stage[wave][s][0]) + lbyte;
        asm volatile("global_load_async_to_lds_b128 %0, %1, off"
                     :: "v"(la), "v"(ga) : "memory");
    }

    // ---- Rows 0..3: regular VMEM path (global_load_b128 -> VGPR, LOADcnt),
    // accumulating while the async engine fills LDS in parallel.
    float4 acc = make_float4(0.f, 0.f, 0.f, 0.f);
#pragma unroll
    for (int s = 0; s < DIRECT; ++s) {
        const float4 v = ((const float4*)(emb + (size_t)id[s] * DFEAT))[lane];
        acc.x += v.x; acc.y += v.y; acc.z += v.z; acc.w += v.w;
    }

    // Drain the async engine (split-counter wait, gfx1250).
    asm volatile("s_wait_asynccnt 0" ::: "memory");

    // ---- Accumulate the staged rows out of LDS (ds_load_b128).
#pragma unroll
    for (int s = 0; s < LDSROWS; ++s) {
        const float4 v = *(const float4*)(&stage[wave][s][lane << 2]);
        acc.x += v.x; acc.y += v.y; acc.z += v.z; acc.w += v.w;
    }

    const float inv = 1.0f / (float)SNEIGH;
    const float4 r = make_float4(acc.x * inv, acc.y * inv, acc.z * inv, acc.w * inv);
    *(float4*)(out + (size_t)node * DFEAT + (lane << 2)) = r;
}

extern "C" void kernel_launch(void* const* d_in, const int* in_sizes, int n_in,
                              void* d_out, int out_size, void* d_ws, size_t ws_size,
                              hipStream_t stream)
{
    const long long* neigh_idx = (const long long*)d_in[0];   // [N, 10] int64
    const float*     emb       = (const float*)d_in[1];       // [200000, 128] f32
    float*           out       = (float*)d_out;               // [N, 128] f32

    const int N = in_sizes[0] / SNEIGH;
    const int blocks = (N + WPB - 1) / WPB;

    mean_agg_gather_kernel<<<blocks, 256, 0, stream>>>(neigh_idx, emb, out, N);
}